// TieCommAgent_2224793060052
// MI455X (gfx1250) — compile-verified
//
#include <hip/hip_runtime.h>
#include <math.h>

#define NA 100000
#define OBSD 256
#define HIDD 128
#define NE 1600000
#define NG 1024
#define OUTW 384   // concat [x(128) | inter(128) | intra(128)]

typedef float v2f __attribute__((ext_vector_type(2)));
typedef float v8f __attribute__((ext_vector_type(8)));

__device__ __forceinline__ v8f wmma_f32(v2f a, v2f b, v8f c) {
    // D = A(16x4) * B(4x16) + C(16x16), all f32
    return __builtin_amdgcn_wmma_f32_16x16x4_f32(false, a, false, b, (short)0, c, false, false);
}

// ---- async global->LDS copy (CDNA5 GLOBAL_LOAD_ASYNC_TO_LDS_B128, ASYNCcnt) ----
#if defined(__has_builtin)
#  if __has_builtin(__builtin_amdgcn_global_load_async_to_lds_b128)
#    define HAVE_ASYNC_LDS 1
#  endif
#endif

typedef int v4i_g __attribute__((vector_size(4 * sizeof(int))));
typedef __attribute__((address_space(1))) v4i_g* gv4_ptr;   // global v4i*
typedef __attribute__((address_space(3))) v4i_g* lv4_ptr;   // LDS v4i*

__device__ __forceinline__ void copy16_to_lds(const float* __restrict__ g, float* l)
{
#ifdef HAVE_ASYNC_LDS
    __builtin_amdgcn_global_load_async_to_lds_b128(
        (gv4_ptr)(void*)(const void*)g, (lv4_ptr)(void*)l, 0, 0);
#else
    *(float4*)l = *(const float4*)g;
#endif
}

__device__ __forceinline__ void async_lds_wait()
{
#ifdef HAVE_ASYNC_LDS
#  if __has_builtin(__builtin_amdgcn_s_wait_asynccnt)
    __builtin_amdgcn_s_wait_asynccnt(0);
#  else
    asm volatile("s_wait_asynccnt 0" ::: "memory");
#  endif
#endif
}

// ---------------------------------------------------------------------------
// K1: x = tanh(obs @ W_emb^T + b_emb)  -> out[:, 0:128]
// block = 256 threads (8 waves). A tile (16 x 256) staged once into LDS,
// each wave computes one 16x16 tile -> block covers 16 rows x 128 cols.
#define LDA_E 260   // 256 + 4 pad floats: bank offset 4 per row, conflict-free b64 reads
__global__ void emb_kernel(const float* __restrict__ obs, const float* __restrict__ Wemb,
                           const float* __restrict__ bemb, float* __restrict__ out)
{
    __shared__ float As[16 * LDA_E];

    const int t    = threadIdx.x;
    const int wave = t >> 5;
    const int lane = t & 31;
    const int row0 = blockIdx.x << 4;
    const int col0 = wave << 4;
    const int mi   = lane & 15;
    const int kq   = (lane >> 4) << 1;          // 0 or 2

    // cooperative stage: 16 rows x 256 f32; thread t -> row t/16, 16 floats
    {
        const int r    = t >> 4;
        const int cseg = (t & 15) << 4;
        const float* g = obs + (size_t)(row0 + r) * OBSD + cseg;
        float*       l = As + r * LDA_E + cseg;
#pragma unroll
        for (int i = 0; i < 4; ++i) copy16_to_lds(g + 4 * i, l + 4 * i);
    }
    async_lds_wait();
    __syncthreads();

    const float* arow = As   + mi * LDA_E;                   // A row (M = mi) in LDS
    const float* brow = Wemb + (size_t)(col0 + mi) * OBSD;   // B col (N = mi), W row-major

    v8f acc = {};
#pragma unroll 8
    for (int k = 0; k < OBSD; k += 4) {
        v2f a, b;
        a.x = arow[k + kq]; a.y = arow[k + kq + 1];
        b.x = brow[k + kq]; b.y = brow[k + kq + 1];
        acc = wmma_f32(a, b, acc);
    }
    const int   mbase = (lane < 16) ? 0 : 8;
    const float bias  = bemb[col0 + mi];
#pragma unroll
    for (int j = 0; j < 8; ++j)
        out[(size_t)(row0 + mbase + j) * OUTW + (col0 + mi)] = tanhf(acc[j] + bias);
}

// ---------------------------------------------------------------------------
// K2: xp = x @ W_gat^T   (x read from out[:,0:128], stride OUTW; A tile in LDS)
#define LDA_G 132   // 128 + 4 pad
__global__ void gat_proj_kernel(const float* __restrict__ out, const float* __restrict__ Wgat,
                                float* __restrict__ xp)
{
    __shared__ float As[16 * LDA_G];

    const int t    = threadIdx.x;
    const int wave = t >> 5;
    const int lane = t & 31;
    const int row0 = blockIdx.x << 4;
    const int col0 = wave << 4;
    const int mi   = lane & 15;
    const int kq   = (lane >> 4) << 1;

    // stage 16 rows x 128 f32 of x (strided in out); thread t -> row t/16, 8 floats
    {
        const int r    = t >> 4;
        const int cseg = (t & 15) << 3;
        const float* g = out + (size_t)(row0 + r) * OUTW + cseg;
        float*       l = As + r * LDA_G + cseg;
        copy16_to_lds(g,     l);
        copy16_to_lds(g + 4, l + 4);
    }
    async_lds_wait();
    __syncthreads();

    const float* arow = As   + mi * LDA_G;
    const float* brow = Wgat + (size_t)(col0 + mi) * HIDD;

    v8f acc = {};
#pragma unroll 8
    for (int k = 0; k < HIDD; k += 4) {
        v2f a, b;
        a.x = arow[k + kq]; a.y = arow[k + kq + 1];
        b.x = brow[k + kq]; b.y = brow[k + kq + 1];
        acc = wmma_f32(a, b, acc);
    }
    const int mbase = (lane < 16) ? 0 : 8;
#pragma unroll
    for (int j = 0; j < 8; ++j)
        xp[(size_t)(row0 + mbase + j) * HIDD + (col0 + mi)] = acc[j];
}

// ---------------------------------------------------------------------------
// K3: a_s = xp @ att_src ; a_d = xp @ att_dst   (wave per row)
__global__ void asad_kernel(const float* __restrict__ xp, const float* __restrict__ att_src,
                            const float* __restrict__ att_dst,
                            float* __restrict__ a_s, float* __restrict__ a_d)
{
    const int wave = threadIdx.x >> 5;
    const int lane = threadIdx.x & 31;
    const int row  = blockIdx.x * 8 + wave;
    const float4 xv = ((const float4*)(xp + (size_t)row * HIDD))[lane];
    const float4 s4 = ((const float4*)att_src)[lane];
    const float4 d4 = ((const float4*)att_dst)[lane];
    float vs = xv.x*s4.x + xv.y*s4.y + xv.z*s4.z + xv.w*s4.w;
    float vd = xv.x*d4.x + xv.y*d4.y + xv.z*d4.z + xv.w*d4.w;
#pragma unroll
    for (int off = 16; off >= 1; off >>= 1) {
        vs += __shfl_xor(vs, off, 32);
        vd += __shfl_xor(vd, off, 32);
    }
    if (lane == 0) { a_s[row] = vs; a_d[row] = vd; }
}

__global__ void fill_neginf_kernel(float* __restrict__ p, int n)
{
    int i = blockIdx.x * blockDim.x + threadIdx.x;
    if (i < n) p[i] = -INFINITY;
}

__device__ __forceinline__ float lrelu02(float v) { return v >= 0.f ? v : 0.2f * v; }

// ---------------------------------------------------------------------------
// K4: segment max of e over dst (int/uint ordered atomic trick, init = -inf)
__global__ void edge_max_kernel(const int* __restrict__ src, const int* __restrict__ dst,
                                const float* __restrict__ a_s, const float* __restrict__ a_d,
                                float* __restrict__ mbuf)
{
    const int e = blockIdx.x * blockDim.x + threadIdx.x;   // NE divisible by 256
    const int s = src[e], d = dst[e];
    const float val = lrelu02(a_s[s] + a_d[d]);
    if (val >= 0.f) atomicMax((int*)(mbuf + d), __float_as_int(val));
    else            atomicMin((unsigned int*)(mbuf + d), __float_as_uint(val));
}

// K5: ex = exp(e - m[dst]); denom[dst] += ex
__global__ void edge_exp_kernel(const int* __restrict__ src, const int* __restrict__ dst,
                                const float* __restrict__ a_s, const float* __restrict__ a_d,
                                const float* __restrict__ mbuf,
                                float* __restrict__ ex, float* __restrict__ denom)
{
    const int e = blockIdx.x * blockDim.x + threadIdx.x;
    const int s = src[e], d = dst[e];
    const float val = lrelu02(a_s[s] + a_d[d]);
    const float ev  = expf(val - mbuf[d]);
    ex[e] = ev;
    atomicAdd(denom + d, ev);
}

// K6: agg[dst] += (ex/denom[dst]) * xp[src]   (wave per edge, 4 floats per lane)
__global__ void edge_scatter_kernel(const int* __restrict__ src, const int* __restrict__ dst,
                                    const float* __restrict__ ex, const float* __restrict__ denom,
                                    const float* __restrict__ xp, float* __restrict__ agg)
{
    const int gw   = (blockIdx.x * blockDim.x + threadIdx.x) >> 5;  // edge id
    const int lane = threadIdx.x & 31;
    const int s = src[gw], d = dst[gw];
    const float w = ex[gw] / fmaxf(denom[d], 1e-16f);
    const float4 xv = ((const float4*)(xp + (size_t)s * HIDD))[lane];
    float* ag = agg + (size_t)d * HIDD + lane * 4;
    atomicAdd(ag + 0, w * xv.x);
    atomicAdd(ag + 1, w * xv.y);
    atomicAdd(ag + 2, w * xv.z);
    atomicAdd(ag + 3, w * xv.w);
}

// K7: intra = tanh(agg + b_gat) -> out[:, 256:384]
__global__ void intra_kernel(const float* __restrict__ agg, const float* __restrict__ bgat,
                             float* __restrict__ out)
{
    const int t    = blockIdx.x * blockDim.x + threadIdx.x;
    const int lane = t & 31;
    const int row  = t >> 5;
    const float4 a = ((const float4*)(agg + (size_t)row * HIDD))[lane];
    const float4 b = ((const float4*)bgat)[lane];
    float4 r;
    r.x = tanhf(a.x + b.x); r.y = tanhf(a.y + b.y);
    r.z = tanhf(a.z + b.z); r.w = tanhf(a.w + b.w);
    ((float4*)(out + (size_t)row * OUTW + 256))[lane] = r;
}

// K8: core[g] = intra[core_node[g]]
__global__ void core_gather_kernel(const float* __restrict__ out, const int* __restrict__ core_node,
                                   float* __restrict__ core)
{
    const int t    = blockIdx.x * blockDim.x + threadIdx.x;
    const int lane = t & 31;
    const int g    = t >> 5;
    const int node = core_node[g];
    ((float4*)(core + (size_t)g * HIDD))[lane] =
        ((const float4*)(out + (size_t)node * OUTW + 256))[lane];
}

// K9: q/k/v = core @ W^T + b  (one wave -> 16x16 tile of each of q,k,v)
__global__ void qkv_kernel(const float* __restrict__ core,
                           const float* __restrict__ Wq, const float* __restrict__ bq,
                           const float* __restrict__ Wk, const float* __restrict__ bk,
                           const float* __restrict__ Wv, const float* __restrict__ bv,
                           float* __restrict__ q, float* __restrict__ k, float* __restrict__ v)
{
    const int wave = threadIdx.x >> 5;
    const int lane = threadIdx.x & 31;
    const int row0 = blockIdx.x << 4;
    const int col0 = wave << 4;
    const int mi   = lane & 15;
    const int kq   = (lane >> 4) << 1;

    const float* arow = core + (size_t)(row0 + mi) * HIDD;
    const float* bqr  = Wq   + (size_t)(col0 + mi) * HIDD;
    const float* bkr  = Wk   + (size_t)(col0 + mi) * HIDD;
    const float* bvr  = Wv   + (size_t)(col0 + mi) * HIDD;

    v8f aq = {}, ak = {}, av = {};
#pragma unroll 4
    for (int kk = 0; kk < HIDD; kk += 4) {
        v2f a, b;
        a.x = arow[kk + kq]; a.y = arow[kk + kq + 1];
        b.x = bqr[kk + kq];  b.y = bqr[kk + kq + 1];
        aq = wmma_f32(a, b, aq);
        b.x = bkr[kk + kq];  b.y = bkr[kk + kq + 1];
        ak = wmma_f32(a, b, ak);
        b.x = bvr[kk + kq];  b.y = bvr[kk + kq + 1];
        av = wmma_f32(a, b, av);
    }
    const int   mbase = (lane < 16) ? 0 : 8;
    const float biq = bq[col0 + mi], bik = bk[col0 + mi], biv = bv[col0 + mi];
#pragma unroll
    for (int j = 0; j < 8; ++j) {
        const size_t o = (size_t)(row0 + mbase + j) * HIDD + (col0 + mi);
        q[o] = aq[j] + biq;
        k[o] = ak[j] + bik;
        v[o] = av[j] + biv;
    }
}

// K10: scores = (q @ k^T) / sqrt(HID)   [1024 x 1024]
__global__ void scores_kernel(const float* __restrict__ q, const float* __restrict__ k,
                              float* __restrict__ scores)
{
    const int wave = threadIdx.x >> 5;
    const int lane = threadIdx.x & 31;
    const int row0 = blockIdx.x << 4;
    const int col0 = blockIdx.y * 128 + (wave << 4);
    const int mi   = lane & 15;
    const int kq   = (lane >> 4) << 1;

    const float* arow = q + (size_t)(row0 + mi) * HIDD;
    const float* brow = k + (size_t)(col0 + mi) * HIDD;

    v8f acc = {};
#pragma unroll 8
    for (int kk = 0; kk < HIDD; kk += 4) {
        v2f a, b;
        a.x = arow[kk + kq]; a.y = arow[kk + kq + 1];
        b.x = brow[kk + kq]; b.y = brow[kk + kq + 1];
        acc = wmma_f32(a, b, acc);
    }
    const int   mbase = (lane < 16) ? 0 : 8;
    const float scale = 0.08838834764831845f;   // 1/sqrt(128)
#pragma unroll
    for (int j = 0; j < 8; ++j)
        scores[(size_t)(row0 + mbase + j) * NG + (col0 + mi)] = acc[j] * scale;
}

// K11: row softmax over scores (wave per row; 1024 elems = 8 float4 per lane)
__global__ void softmax_kernel(float* __restrict__ scores)
{
    const int wave = threadIdx.x >> 5;
    const int lane = threadIdx.x & 31;
    const int row  = blockIdx.x * 8 + wave;
    float4* rp = (float4*)(scores + (size_t)row * NG);

    float4 buf[8];
    float mx = -INFINITY;
#pragma unroll
    for (int i = 0; i < 8; ++i) {
        buf[i] = rp[lane + 32 * i];
        mx = fmaxf(mx, fmaxf(fmaxf(buf[i].x, buf[i].y), fmaxf(buf[i].z, buf[i].w)));
    }
#pragma unroll
    for (int off = 16; off >= 1; off >>= 1) mx = fmaxf(mx, __shfl_xor(mx, off, 32));

    float sum = 0.f;
#pragma unroll
    for (int i = 0; i < 8; ++i) {
        buf[i].x = expf(buf[i].x - mx); buf[i].y = expf(buf[i].y - mx);
        buf[i].z = expf(buf[i].z - mx); buf[i].w = expf(buf[i].w - mx);
        sum += buf[i].x + buf[i].y + buf[i].z + buf[i].w;
    }
#pragma unroll
    for (int off = 16; off >= 1; off >>= 1) sum += __shfl_xor(sum, off, 32);

    const float inv = 1.f / sum;
#pragma unroll
    for (int i = 0; i < 8; ++i) {
        buf[i].x *= inv; buf[i].y *= inv; buf[i].z *= inv; buf[i].w *= inv;
        rp[lane + 32 * i] = buf[i];
    }
}

// K12: av = attn @ v   ([1024,1024] @ [1024,128])
__global__ void attnv_kernel(const float* __restrict__ scores, const float* __restrict__ v,
                             float* __restrict__ av)
{
    const int wave = threadIdx.x >> 5;
    const int lane = threadIdx.x & 31;
    const int row0 = blockIdx.x << 4;
    const int col0 = wave << 4;
    const int mi   = lane & 15;
    const int kq   = (lane >> 4) << 1;

    const float* arow = scores + (size_t)(row0 + mi) * NG;

    v8f acc = {};
#pragma unroll 4
    for (int kk = 0; kk < NG; kk += 4) {
        v2f a, b;
        a.x = arow[kk + kq]; a.y = arow[kk + kq + 1];
        b.x = v[(size_t)(kk + kq)     * HIDD + (col0 + mi)];
        b.y = v[(size_t)(kk + kq + 1) * HIDD + (col0 + mi)];
        acc = wmma_f32(a, b, acc);
    }
    const int mbase = (lane < 16) ? 0 : 8;
#pragma unroll
    for (int j = 0; j < 8; ++j)
        av[(size_t)(row0 + mbase + j) * HIDD + (col0 + mi)] = acc[j];
}

// K13: go = av @ Wo^T + bo
__global__ void go_kernel(const float* __restrict__ av, const float* __restrict__ Wo,
                          const float* __restrict__ bo, float* __restrict__ go)
{
    const int wave = threadIdx.x >> 5;
    const int lane = threadIdx.x & 31;
    const int row0 = blockIdx.x << 4;
    const int col0 = wave << 4;
    const int mi   = lane & 15;
    const int kq   = (lane >> 4) << 1;

    const float* arow = av + (size_t)(row0 + mi) * HIDD;
    const float* brow = Wo + (size_t)(col0 + mi) * HIDD;

    v8f acc = {};
#pragma unroll 8
    for (int kk = 0; kk < HIDD; kk += 4) {
        v2f a, b;
        a.x = arow[kk + kq]; a.y = arow[kk + kq + 1];
        b.x = brow[kk + kq]; b.y = brow[kk + kq + 1];
        acc = wmma_f32(a, b, acc);
    }
    const int   mbase = (lane < 16) ? 0 : 8;
    const float bias  = bo[col0 + mi];
#pragma unroll
    for (int j = 0; j < 8; ++j)
        go[(size_t)(row0 + mbase + j) * HIDD + (col0 + mi)] = acc[j] + bias;
}

// K14: out[n, 128:256] = go[group_ids[n]]
__global__ void inter_scatter_kernel(const float* __restrict__ go, const int* __restrict__ gid,
                                     float* __restrict__ out)
{
    const int t    = blockIdx.x * blockDim.x + threadIdx.x;
    const int lane = t & 31;
    const int row  = t >> 5;
    const int g    = gid[row];
    ((float4*)(out + (size_t)row * OUTW + 128))[lane] =
        ((const float4*)(go + (size_t)g * HIDD))[lane];
}

// ---------------------------------------------------------------------------
extern "C" void kernel_launch(void* const* d_in, const int* in_sizes, int n_in,
                              void* d_out, int out_size, void* d_ws, size_t ws_size,
                              hipStream_t stream)
{
    const float* obs      = (const float*)d_in[0];
    const float* Wemb     = (const float*)d_in[1];
    const float* bemb     = (const float*)d_in[2];
    const float* Wgat     = (const float*)d_in[3];
    const float* att_src  = (const float*)d_in[4];
    const float* att_dst  = (const float*)d_in[5];
    const float* bgat     = (const float*)d_in[6];
    const float* Wq       = (const float*)d_in[7];
    const float* bq       = (const float*)d_in[8];
    const float* Wk       = (const float*)d_in[9];
    const float* bk       = (const float*)d_in[10];
    const float* Wv       = (const float*)d_in[11];
    const float* bv       = (const float*)d_in[12];
    const float* Wo       = (const float*)d_in[13];
    const float* bo       = (const float*)d_in[14];
    const int*   edge     = (const int*)d_in[15];
    const int*   gid      = (const int*)d_in[16];
    const int*   corenode = (const int*)d_in[17];
    float* out = (float*)d_out;

    const int* src = edge;
    const int* dst = edge + NE;

    // workspace layout (floats)
    float* ws     = (float*)d_ws;
    float* xp     = ws;                         // NA*HID
    float* agg    = xp     + (size_t)NA * HIDD; // NA*HID
    float* a_s    = agg    + (size_t)NA * HIDD; // NA
    float* a_d    = a_s    + NA;                // NA
    float* mbuf   = a_d    + NA;                // NA
    float* denom  = mbuf   + NA;                // NA
    float* exbuf  = denom  + NA;                // NE
    float* core   = exbuf  + NE;                // NG*HID
    float* qb     = core   + (size_t)NG * HIDD;
    float* kb     = qb     + (size_t)NG * HIDD;
    float* vb     = kb     + (size_t)NG * HIDD;
    float* scores = vb     + (size_t)NG * HIDD; // NG*NG
    float* avb    = scores + (size_t)NG * NG;
    float* gob    = avb    + (size_t)NG * HIDD;

    // init accumulators (graph-capturable async memsets + fill kernel)
    (void)hipMemsetAsync(agg,   0, (size_t)NA * HIDD * sizeof(float), stream);
    (void)hipMemsetAsync(denom, 0, (size_t)NA * sizeof(float), stream);
    fill_neginf_kernel<<<(NA + 255) / 256, 256, 0, stream>>>(mbuf, NA);

    // 1) embedding GEMM + tanh -> out[:,0:128]
    emb_kernel<<<NA / 16, 256, 0, stream>>>(obs, Wemb, bemb, out);
    // 2) xp = x @ Wgat^T
    gat_proj_kernel<<<NA / 16, 256, 0, stream>>>(out, Wgat, xp);
    // 3) a_s, a_d
    asad_kernel<<<NA / 8, 256, 0, stream>>>(xp, att_src, att_dst, a_s, a_d);
    // 4) segment max
    edge_max_kernel<<<NE / 256, 256, 0, stream>>>(src, dst, a_s, a_d, mbuf);
    // 5) exp + denom
    edge_exp_kernel<<<NE / 256, 256, 0, stream>>>(src, dst, a_s, a_d, mbuf, exbuf, denom);
    // 6) weighted scatter (wave per edge)
    edge_scatter_kernel<<<NE / 8, 256, 0, stream>>>(src, dst, exbuf, denom, xp, agg);
    // 7) intra = tanh(agg + b) -> out[:,256:384]
    intra_kernel<<<(NA * 32) / 256, 256, 0, stream>>>(agg, bgat, out);
    // 8) core gather
    core_gather_kernel<<<(NG * 32) / 256, 256, 0, stream>>>(out, corenode, core);
    // 9) q,k,v
    qkv_kernel<<<NG / 16, 256, 0, stream>>>(core, Wq, bq, Wk, bk, Wv, bv, qb, kb, vb);
    // 10) scores
    scores_kernel<<<dim3(NG / 16, NG / 128), 256, 0, stream>>>(qb, kb, scores);
    // 11) softmax
    softmax_kernel<<<NG / 8, 256, 0, stream>>>(scores);
    // 12) attn @ v
    attnv_kernel<<<NG / 16, 256, 0, stream>>>(scores, vb, avb);
    // 13) go = av @ Wo^T + bo
    go_kernel<<<NG / 16, 256, 0, stream>>>(avb, Wo, bo, gob);
    // 14) scatter inter -> out[:,128:256]
    inter_scatter_kernel<<<(NA * 32) / 256, 256, 0, stream>>>(gob, gid, out);
}